// GAE_23356032156160
// MI455X (gfx1250) — compile-verified
//
#include <hip/hip_runtime.h>
#include <hip/hip_bf16.h>

typedef __attribute__((ext_vector_type(2))) float v2f;
typedef __attribute__((ext_vector_type(8))) float v8f;

#define GAE_N_NODES 8192
#define GAE_N_EDGES 262144
#define GAE_F_IN    512

// ---------------------------------------------------------------------------
// GEMM1: HW1[8192x16] = X[8192x512] @ W1[512x16], f32 WMMA 16x16x4.
// One wave computes one 16(M) x 16(N) tile; 512 waves total.
// A layout (16x4 f32): lanes 0-15 hold K={0,1} in v0/v1, lanes 16-31 K={2,3}.
// B layout (4x16 f32): lanes 0-15 hold rows K={0,1}, lanes 16-31 rows K={2,3}.
// ---------------------------------------------------------------------------
__global__ void __launch_bounds__(256)
gae_gemm1_xw1(const float* __restrict__ X, const float* __restrict__ W1,
              float* __restrict__ HW1) {
    __shared__ float sW[GAE_F_IN * 16];   // 32 KB of the 320 KB LDS
    for (int i = threadIdx.x; i < GAE_F_IN * 16; i += 256) sW[i] = W1[i];
    __syncthreads();

    const int wave = (blockIdx.x * blockDim.x + threadIdx.x) >> 5;  // 0..511
    const int lane = threadIdx.x & 31;
    const int half = lane >> 4;        // 0: K lo pair, 1: K hi pair
    const int l    = lane & 15;
    const int m0   = wave * 16;

    const float* arow = X + (size_t)(m0 + l) * GAE_F_IN;
    v8f c = {};
    #pragma unroll 4
    for (int k = 0; k < GAE_F_IN; k += 4) {
        v2f a = *(const v2f*)(arow + k + 2 * half);          // b64 load
        v2f b;
        b.x = sW[(k + 2 * half) * 16 + l];
        b.y = sW[(k + 2 * half + 1) * 16 + l];
        c = __builtin_amdgcn_wmma_f32_16x16x4_f32(false, a, false, b,
                                                  (short)0, c, false, false);
    }
    // C/D layout: VGPR v -> row M = v + 8*half, col N = l
    #pragma unroll
    for (int v = 0; v < 8; ++v)
        HW1[(size_t)(m0 + v + 8 * half) * 16 + l] = c[v];
}

// ---------------------------------------------------------------------------
// Edge aggregation, 16 features: dst[row][f] += w * src[col][f]
// ---------------------------------------------------------------------------
__global__ void __launch_bounds__(256)
gae_edge_agg16(const int* __restrict__ erow, const int* __restrict__ ecol,
               const float* __restrict__ ew, const float* __restrict__ src,
               float* __restrict__ dst) {
    const int idx = blockIdx.x * blockDim.x + threadIdx.x;  // N_EDGES*16
    const int e = idx >> 4;
    const int f = idx & 15;
    const int r = erow[e];
    const int cc = ecol[e];
    const float v = ew[e] * src[(size_t)cc * 16 + f];
    atomicAdd(&dst[(size_t)r * 16 + f], v);
}

// H = relu(H + b1), 16-wide
__global__ void __launch_bounds__(256)
gae_bias_relu16(float* __restrict__ H, const float* __restrict__ b1) {
    const int idx = blockIdx.x * blockDim.x + threadIdx.x;  // N_NODES*16
    const float v = H[idx] + b1[idx & 15];
    H[idx] = v > 0.0f ? v : 0.0f;
}

// ---------------------------------------------------------------------------
// GEMM2: HW2pad[8192x8] = H1[8192x16] @ W2[16x7] (N padded to 16 with zeros,
// only cols 0..7 stored; col 7 computes to 0 -> pad column is zeroed for free).
// ---------------------------------------------------------------------------
__global__ void __launch_bounds__(256)
gae_gemm2_hw2(const float* __restrict__ H1, const float* __restrict__ W2,
              float* __restrict__ HW2) {
    const int wave = (blockIdx.x * blockDim.x + threadIdx.x) >> 5;  // 0..511
    const int lane = threadIdx.x & 31;
    const int half = lane >> 4;
    const int l    = lane & 15;
    const int m0   = wave * 16;

    v8f c = {};
    #pragma unroll
    for (int k = 0; k < 16; k += 4) {
        v2f a = *(const v2f*)(H1 + (size_t)(m0 + l) * 16 + k + 2 * half);
        v2f b;
        b.x = (l < 7) ? W2[(k + 2 * half) * 7 + l] : 0.0f;
        b.y = (l < 7) ? W2[(k + 2 * half + 1) * 7 + l] : 0.0f;
        c = __builtin_amdgcn_wmma_f32_16x16x4_f32(false, a, false, b,
                                                  (short)0, c, false, false);
    }
    #pragma unroll
    for (int v = 0; v < 8; ++v)
        if (l < 8)
            HW2[(size_t)(m0 + v + 8 * half) * 8 + l] = c[v];
}

// Edge aggregation, 8-wide padded (pad lane f=7 stays exactly 0.0)
__global__ void __launch_bounds__(256)
gae_edge_agg8(const int* __restrict__ erow, const int* __restrict__ ecol,
              const float* __restrict__ ew, const float* __restrict__ src,
              float* __restrict__ dst) {
    const int idx = blockIdx.x * blockDim.x + threadIdx.x;  // N_EDGES*8
    const int e = idx >> 3;
    const int f = idx & 7;
    const int r = erow[e];
    const int cc = ecol[e];
    const float v = ew[e] * src[(size_t)cc * 8 + f];
    atomicAdd(&dst[(size_t)r * 8 + f], v);
}

// Z += b2 (cols 0..6 only; pad column 7 untouched / zero)
__global__ void __launch_bounds__(256)
gae_bias8(float* __restrict__ Z, const float* __restrict__ b2) {
    const int idx = blockIdx.x * blockDim.x + threadIdx.x;  // N_NODES*8
    const int f = idx & 7;
    if (f < 7) Z[idx] += b2[f];
}

// ---------------------------------------------------------------------------
// Decoder: out[8192*8192] = Z @ Z^T, Z padded to [8192 x 8] f32.
// One wave per 16x16 output tile (512 x 512 tiles). K=8 -> two WMMA K=4 steps.
// Each lane loads its full 8-float Z row with two b128 loads, then slices the
// K pair it owns per the A/B VGPR layouts. Z is 256 KB -> L2 resident; the
// kernel is bounded by the 256 MB streaming store (~11 us at 23.3 TB/s).
// Output is write-once and exceeds the 192 MB L2, so stores use the
// non-temporal hint (TH=NT) to keep the write stream from evicting Z.
// ---------------------------------------------------------------------------
__global__ void __launch_bounds__(256)
gae_decoder_zzt(const float* __restrict__ Z, float* __restrict__ out) {
    const int wave = (blockIdx.x * blockDim.x + threadIdx.x) >> 5;  // 0..262143
    const int lane = threadIdx.x & 31;
    const int half = lane >> 4;
    const int l    = lane & 15;
    const int r0 = (wave >> 9) * 16;
    const int c0 = (wave & 511) * 16;

    const float4* za = (const float4*)(Z + (size_t)(r0 + l) * 8);
    const float4* zb = (const float4*)(Z + (size_t)(c0 + l) * 8);
    const float4 qa0 = za[0], qa1 = za[1];     // K 0..3, 4..7 of row r0+l
    const float4 qb0 = zb[0], qb1 = zb[1];     // K 0..3, 4..7 of row c0+l

    v8f c = {};
    v2f a, b;
    // K-step 0 (K=0..3): lanes 0-15 take K{0,1}, lanes 16-31 take K{2,3}
    a.x = half ? qa0.z : qa0.x;  a.y = half ? qa0.w : qa0.y;
    b.x = half ? qb0.z : qb0.x;  b.y = half ? qb0.w : qb0.y;
    c = __builtin_amdgcn_wmma_f32_16x16x4_f32(false, a, false, b,
                                              (short)0, c, false, false);
    // K-step 1 (K=4..7; col 7 of Z is the zero pad)
    a.x = half ? qa1.z : qa1.x;  a.y = half ? qa1.w : qa1.y;
    b.x = half ? qb1.z : qb1.x;  b.y = half ? qb1.w : qb1.y;
    c = __builtin_amdgcn_wmma_f32_16x16x4_f32(false, a, false, b,
                                              (short)0, c, false, false);

    float* orow = out + (size_t)(r0 + 8 * half) * GAE_N_NODES + (c0 + l);
    #pragma unroll
    for (int v = 0; v < 8; ++v)
        __builtin_nontemporal_store(c[v], orow + (size_t)v * GAE_N_NODES);
}

// ---------------------------------------------------------------------------
extern "C" void kernel_launch(void* const* d_in, const int* in_sizes, int n_in,
                              void* d_out, int out_size, void* d_ws, size_t ws_size,
                              hipStream_t stream) {
    const float* X    = (const float*)d_in[0];
    const int*   erow = (const int*)  d_in[1];
    const int*   ecol = (const int*)  d_in[2];
    const float* ew   = (const float*)d_in[3];
    const float* W1   = (const float*)d_in[4];
    const float* b1   = (const float*)d_in[5];
    const float* W2   = (const float*)d_in[6];
    const float* b2   = (const float*)d_in[7];
    float* out = (float*)d_out;

    float* ws  = (float*)d_ws;
    float* HW1 = ws;                          // 8192*16 = 131072 f32 (512 KB)
    float* H1  = ws + 131072;                 // 131072 f32 (512 KB)
    float* HW2 = ws + 262144;                 // 8192*8 = 65536 f32 (256 KB)
    float* Zp  = ws + 262144 + 65536;         // 65536 f32 (256 KB)

    // Layer 1: X@W1 -> sparse agg -> +b1, relu
    gae_gemm1_xw1<<<64, 256, 0, stream>>>(X, W1, HW1);                 // 512 waves
    hipMemsetAsync(H1, 0, 131072 * sizeof(float), stream);
    gae_edge_agg16<<<(GAE_N_EDGES * 16) / 256, 256, 0, stream>>>(erow, ecol, ew, HW1, H1);
    gae_bias_relu16<<<(GAE_N_NODES * 16) / 256, 256, 0, stream>>>(H1, b1);

    // Layer 2: H1@W2 -> sparse agg -> +b2  (8-wide zero-padded Z)
    gae_gemm2_hw2<<<64, 256, 0, stream>>>(H1, W2, HW2);
    hipMemsetAsync(Zp, 0, 65536 * sizeof(float), stream);
    gae_edge_agg8<<<(GAE_N_EDGES * 8) / 256, 256, 0, stream>>>(erow, ecol, ew, HW2, Zp);
    gae_bias8<<<(GAE_N_NODES * 8) / 256, 256, 0, stream>>>(Zp, b2);

    // Decoder: out = Z @ Z^T  (262144 tiles, 8 waves/block)
    gae_decoder_zzt<<<32768, 256, 0, stream>>>(Zp, out);
}